// ToroidalSOM_85040352461292
// MI455X (gfx1250) — compile-verified
//
#include <hip/hip_runtime.h>
#include <hip/hip_bf16.h>

// Problem constants from the reference
#define SOM_B     128
#define SOM_D     64
#define SOM_N     16384          // ROWS*COLS = 128*128
#define NTILES    (SOM_N / 16)   // 1024
#define BTILES    (SOM_B / 16)   // 8

typedef __attribute__((ext_vector_type(2))) float v2f;
typedef __attribute__((ext_vector_type(8))) float v8f;

// One wave32 computes one 16(batch) x 16(neuron) output tile using
// V_WMMA_F32_16X16X4_F32 over the D=64 reduction (16 chunks of K=4).
//
// dist[b,n] = ||x_b||^2 + ||w_n||^2 - 2 * dot(x_b, w_n)
__global__ __launch_bounds__(256)
void som_dist_wmma_f32(const float* __restrict__ x,
                       const float* __restrict__ w,
                       float* __restrict__ out) {
  const int lane  = threadIdx.x & 31;
  const int wvid  = (blockIdx.x << 3) + (threadIdx.x >> 5); // 8 waves/block
  const int ntile = wvid & (NTILES - 1);
  const int btile = wvid >> 10;            // wvid / NTILES

  const int row = lane & 15;               // M for A-loads, N for B-loads
  const int hi  = lane >> 4;               // lane group selects K pair

  // A (16x4 f32) layout: VGPR0 = K{0|2}, VGPR1 = K{1|3}; lanes 0-15 -> K0/K1,
  // lanes 16-31 -> K2/K3.  B (4x16) is symmetric with N = lane%16.
  // Both reduce to: lane loads float2 at k = 4c + 2*hi from its row.
  const float* __restrict__ xrow = x + (size_t)(btile * 16 + row) * SOM_D + 2 * hi;
  const float* __restrict__ wrow = w + (size_t)(ntile * 16 + row) * SOM_D + 2 * hi;

  v8f acc = {};                  // C/D 16x16 f32 accumulator (8 VGPRs)
  float xs = 0.0f;               // partial ||x_row||^2 (this lane's half)
  float ws = 0.0f;               // partial ||w_row||^2 (this lane's half)

#pragma unroll
  for (int c = 0; c < 16; ++c) {
    v2f a = *(const v2f*)(xrow + 4 * c);
    v2f b = *(const v2f*)(wrow + 4 * c);
    xs += a.x * a.x + a.y * a.y;
    ws += b.x * b.x + b.y * b.y;
    // 8 args: (neg_a, A, neg_b, B, c_mod, C, reuse_a, reuse_b); neg must be 0 for f32
    acc = __builtin_amdgcn_wmma_f32_16x16x4_f32(
        false, a, false, b, (short)0, acc, false, false);
  }

  // Combine the two half-row partial norms (lanes l and l^16 hold halves of
  // the same row): after this, lane l holds the full norm of row (l & 15).
  xs += __shfl_xor(xs, 16, 32);
  ws += __shfl_xor(ws, 16, 32);

  // C/D layout: VGPR i, lane l -> element (M = 8*(l>>4) + i, N = l & 15).
  // wn (column norm) is exactly this lane's ws; xn (row norm for M) lives in
  // lane M, fetched with a per-lane bpermute.
  float* __restrict__ obase =
      out + (size_t)(btile * 16 + 8 * hi) * SOM_N + ntile * 16 + row;

#pragma unroll
  for (int i = 0; i < 8; ++i) {
    float xn = __shfl(xs, 8 * hi + i, 32);
    obase[(size_t)i * SOM_N] = xn + ws - 2.0f * acc[i];
  }
}

extern "C" void kernel_launch(void* const* d_in, const int* in_sizes, int n_in,
                              void* d_out, int out_size, void* d_ws, size_t ws_size,
                              hipStream_t stream) {
  (void)in_sizes; (void)n_in; (void)d_ws; (void)ws_size; (void)out_size;
  const float* x = (const float*)d_in[0];   // (128, 64)
  const float* w = (const float*)d_in[1];   // (128, 128, 64) viewed as (16384, 64)
  float* out = (float*)d_out;               // (128, 16384)

  // 8192 tiles -> 8192 waves -> 1024 blocks of 256 threads (8 waves/block)
  dim3 grid(BTILES * NTILES / 8);
  dim3 block(256);
  som_dist_wmma_f32<<<grid, block, 0, stream>>>(x, w, out);
}